// Self_Att_28862180229464
// MI455X (gfx1250) — compile-verified
//
#include <hip/hip_runtime.h>
#include <hip/hip_bf16.h>

// ---------------------------------------------------------------------------
// Self-attention (B=4, C=64, H=W=64 -> N=4096, CQ=8), f32 end-to-end.
// Outputs: out (B*C*N floats) ++ att_t (B*N*N floats) in d_out.
// Bandwidth-bound (512MB att traffic); matrix math via V_WMMA_F32_16X16X4_F32;
// GEMM staging via GLOBAL_LOAD_ASYNC_TO_LDS_B128 when the builtin is available.
// ---------------------------------------------------------------------------

#define B_   4
#define C_   64
#define N_   4096
#define CQ_  8

typedef float v2f __attribute__((ext_vector_type(2)));
typedef float v8f __attribute__((ext_vector_type(8)));
typedef int   v4i_ __attribute__((ext_vector_type(4)));

// D(16x16) = A(16x4) * B(4x16) + C, all f32, wave32.
__device__ __forceinline__ v8f wmma4(v2f a, v2f b, v8f c) {
  return __builtin_amdgcn_wmma_f32_16x16x4_f32(
      /*neg_a=*/false, a, /*neg_b=*/false, b,
      /*c_mod=*/(short)0, c, /*reuse_a=*/false, /*reuse_b=*/false);
}

// --- CDNA5 async global->LDS copy (ASYNCcnt-tracked), with safe fallback ----
#if __has_builtin(__builtin_amdgcn_global_load_async_to_lds_b128)
#define ASYNC_LDS 1
#endif

#ifdef ASYNC_LDS
typedef v4i_ __attribute__((address_space(1))) gv4i;  // global int4
typedef v4i_ __attribute__((address_space(3))) lv4i;  // LDS int4
#endif

__device__ __forceinline__ void async_ld16(float* ldst, const float* gsrc) {
#ifdef ASYNC_LDS
  __builtin_amdgcn_global_load_async_to_lds_b128(
      (gv4i*)(gsrc), (lv4i*)(ldst), /*offset=*/0, /*cpol=*/0);
#else
  *(float4*)ldst = *(const float4*)gsrc;
#endif
}

__device__ __forceinline__ void wait_async() {
#ifdef ASYNC_LDS
#if __has_builtin(__builtin_amdgcn_s_wait_asynccnt)
  __builtin_amdgcn_s_wait_asynccnt(0);
#else
  asm volatile("s_wait_asynccnt 0x0" ::: "memory");
#endif
#endif
}

// ---------------------------------------------------------------------------
// Kernel 1: q,k,v 1x1-conv projections.
//   qw[b][o][n] = sum_c wq[o][c] x[b][c][n] + bq[o]   (o<8)
//   kw same with wk/bk; vw[b][o][n] with wv/bv (o<64)
// One thread per (b,n). Weights staged in LDS.
// ---------------------------------------------------------------------------
__global__ void qkv_kernel(const float* __restrict__ x,
                           const float* __restrict__ wq, const float* __restrict__ bq,
                           const float* __restrict__ wk, const float* __restrict__ bk,
                           const float* __restrict__ wv, const float* __restrict__ bv,
                           float* __restrict__ qwp, float* __restrict__ kwp,
                           float* __restrict__ vwp) {
  __shared__ float swq[CQ_ * C_];
  __shared__ float swk[CQ_ * C_];
  __shared__ float swv[C_ * C_];
  __shared__ float sbq[CQ_], sbk[CQ_], sbv[C_];

  const int t = threadIdx.x;
  for (int i = t; i < CQ_ * C_; i += 256) { swq[i] = wq[i]; swk[i] = wk[i]; }
  for (int i = t; i < C_ * C_;  i += 256) { swv[i] = wv[i]; }
  if (t < CQ_) { sbq[t] = bq[t]; sbk[t] = bk[t]; }
  if (t < C_)  { sbv[t] = bv[t]; }
  __syncthreads();

  const int blocksPerBatch = N_ / 256;
  const int b = blockIdx.x / blocksPerBatch;
  const int n = (blockIdx.x % blocksPerBatch) * 256 + t;

  float qa[CQ_], ka[CQ_], va[C_];
#pragma unroll
  for (int o = 0; o < CQ_; ++o) { qa[o] = sbq[o]; ka[o] = sbk[o]; }
#pragma unroll
  for (int o = 0; o < C_; ++o) { va[o] = sbv[o]; }

  const float* xp = x + (size_t)b * C_ * N_ + n;
  for (int c = 0; c < C_; ++c) {
    const float xv = xp[(size_t)c * N_];
#pragma unroll
    for (int o = 0; o < CQ_; ++o) {
      qa[o] = fmaf(swq[o * C_ + c], xv, qa[o]);
      ka[o] = fmaf(swk[o * C_ + c], xv, ka[o]);
    }
#pragma unroll
    for (int o = 0; o < C_; ++o) {
      va[o] = fmaf(swv[o * C_ + c], xv, va[o]);
    }
  }

  const size_t qbase = (size_t)b * CQ_ * N_ + n;
#pragma unroll
  for (int o = 0; o < CQ_; ++o) {
    qwp[qbase + (size_t)o * N_] = qa[o];
    kwp[qbase + (size_t)o * N_] = ka[o];
  }
  const size_t vbase = (size_t)b * C_ * N_ + n;
#pragma unroll
  for (int o = 0; o < C_; ++o) vwp[vbase + (size_t)o * N_] = va[o];
}

// ---------------------------------------------------------------------------
// Kernel 2a: softmax row statistics (online max/sum) over k for each q row.
// Block = 8 waves, one 16-row q tile; each wave streams 32 k-tiles.
// Logits via 2x wmma4 (K=8). Online (m,l) kept PER-LANE; cross-lane merge
// once per wave at the end (shfl butterfly), then across waves in LDS.
// Stores row max and 1/rowsum (reciprocal precomputed for kernel 2b).
// ---------------------------------------------------------------------------
__global__ void att_stats_kernel(const float* __restrict__ qwp,
                                 const float* __restrict__ kwp,
                                 float* __restrict__ rowm,
                                 float* __restrict__ rowinv) {
  const int b    = blockIdx.y;
  const int q0   = blockIdx.x * 16;
  const int wave = threadIdx.x >> 5;
  const int lane = threadIdx.x & 31;
  const int m    = lane & 15;
  const int hi   = lane >> 4;

  const float* qb = qwp + (size_t)b * CQ_ * N_;
  const float* kb = kwp + (size_t)b * CQ_ * N_;

  // A fragment (q tile), invariant over k: K rows {0,1}/{2,3} then {4,5}/{6,7}.
  v2f aLo, aHi;
  aLo.x = qb[(size_t)(0 + 2 * hi) * N_ + q0 + m];
  aLo.y = qb[(size_t)(1 + 2 * hi) * N_ + q0 + m];
  aHi.x = qb[(size_t)(4 + 2 * hi) * N_ + q0 + m];
  aHi.y = qb[(size_t)(5 + 2 * hi) * N_ + q0 + m];

  float rm[8], rl[8];
#pragma unroll
  for (int i = 0; i < 8; ++i) { rm[i] = -3.0e38f; rl[i] = 0.0f; }

  const int tilesPerWave = (N_ / 16) / 8;  // 32
  int k0 = wave * tilesPerWave * 16;
  for (int tk = 0; tk < tilesPerWave; ++tk, k0 += 16) {
    v2f bLo, bHi;
    bLo.x = kb[(size_t)(0 + 2 * hi) * N_ + k0 + m];
    bLo.y = kb[(size_t)(1 + 2 * hi) * N_ + k0 + m];
    bHi.x = kb[(size_t)(4 + 2 * hi) * N_ + k0 + m];
    bHi.y = kb[(size_t)(5 + 2 * hi) * N_ + k0 + m];

    v8f acc = {0.f, 0.f, 0.f, 0.f, 0.f, 0.f, 0.f, 0.f};
    acc = wmma4(aLo, bLo, acc);
    acc = wmma4(aHi, bHi, acc);

    // Per-lane online update: one element per row, no cross-lane traffic.
#pragma unroll
    for (int i = 0; i < 8; ++i) {
      const float d  = acc[i];
      const float mn = fmaxf(rm[i], d);
      rl[i] = rl[i] * __expf(rm[i] - mn) + __expf(d - mn);
      rm[i] = mn;
    }
  }

  // Merge (m,l) across the 16 lanes of each half (once per wave).
#pragma unroll
  for (int i = 0; i < 8; ++i) {
    float mc = rm[i], lc = rl[i];
#pragma unroll
    for (int mask = 1; mask <= 8; mask <<= 1) {
      const float mo = __shfl_xor(mc, mask, 32);
      const float lo = __shfl_xor(lc, mask, 32);
      const float mn = fmaxf(mc, mo);
      lc = lc * __expf(mc - mn) + lo * __expf(mo - mn);
      mc = mn;
    }
    rm[i] = mc;
    rl[i] = lc;
  }

  // Combine the 8 waves' partial (m,l) per row.
  __shared__ float cm[8][16];
  __shared__ float cl[8][16];
  if ((lane & 15) == 0) {
#pragma unroll
    for (int i = 0; i < 8; ++i) {
      cm[wave][i + 8 * hi] = rm[i];
      cl[wave][i + 8 * hi] = rl[i];
    }
  }
  __syncthreads();
  if (threadIdx.x < 16) {
    const int r = threadIdx.x;
    float M = -3.0e38f;
#pragma unroll
    for (int w = 0; w < 8; ++w) M = fmaxf(M, cm[w][r]);
    float L = 0.0f;
#pragma unroll
    for (int w = 0; w < 8; ++w) L += cl[w][r] * __expf(cm[w][r] - M);
    rowm[(size_t)b * N_ + q0 + r]   = M;
    rowinv[(size_t)b * N_ + q0 + r] = 1.0f / L;
  }
}

// ---------------------------------------------------------------------------
// Kernel 2b: recompute logits, normalize, write att_t[b][k][q] into d_out.
// Each wave handles 4 consecutive k-tiles for one q tile: A-fragment and row
// stats loaded once; per tile: 4 B loads + 2 WMMA + 8 exp + 2 b128 stores.
// ---------------------------------------------------------------------------
__global__ void att_write_kernel(const float* __restrict__ qwp,
                                 const float* __restrict__ kwp,
                                 const float* __restrict__ rowm,
                                 const float* __restrict__ rowinv,
                                 float* __restrict__ att_t) {
  const int b     = blockIdx.z;
  const int q0    = blockIdx.x * 16;
  const int wave  = threadIdx.x >> 5;
  const int lane  = threadIdx.x & 31;
  const int m     = lane & 15;
  const int hi    = lane >> 4;
  const int kbase = (blockIdx.y * 8 + wave) * 64;   // 4 tiles of 16

  const float* qb = qwp + (size_t)b * CQ_ * N_;
  const float* kb = kwp + (size_t)b * CQ_ * N_;

  v2f aLo, aHi;
  aLo.x = qb[(size_t)(0 + 2 * hi) * N_ + q0 + m];
  aLo.y = qb[(size_t)(1 + 2 * hi) * N_ + q0 + m];
  aHi.x = qb[(size_t)(4 + 2 * hi) * N_ + q0 + m];
  aHi.y = qb[(size_t)(5 + 2 * hi) * N_ + q0 + m];

  // Per-lane row stats: rows q0 + 8*hi + [0..7]. rowinv already reciprocal.
  const float4 m01 = *(const float4*)(rowm   + (size_t)b * N_ + q0 + 8 * hi);
  const float4 m23 = *(const float4*)(rowm   + (size_t)b * N_ + q0 + 8 * hi + 4);
  const float4 i01 = *(const float4*)(rowinv + (size_t)b * N_ + q0 + 8 * hi);
  const float4 i23 = *(const float4*)(rowinv + (size_t)b * N_ + q0 + 8 * hi + 4);
  const float mr[8] = {m01.x, m01.y, m01.z, m01.w, m23.x, m23.y, m23.z, m23.w};
  const float ir[8] = {i01.x, i01.y, i01.z, i01.w, i23.x, i23.y, i23.z, i23.w};

#pragma unroll
  for (int tt = 0; tt < 4; ++tt) {
    const int k0 = kbase + tt * 16;

    v2f bLo, bHi;
    bLo.x = kb[(size_t)(0 + 2 * hi) * N_ + k0 + m];
    bLo.y = kb[(size_t)(1 + 2 * hi) * N_ + k0 + m];
    bHi.x = kb[(size_t)(4 + 2 * hi) * N_ + k0 + m];
    bHi.y = kb[(size_t)(5 + 2 * hi) * N_ + k0 + m];

    v8f acc = {0.f, 0.f, 0.f, 0.f, 0.f, 0.f, 0.f, 0.f};
    acc = wmma4(aLo, bLo, acc);
    acc = wmma4(aHi, bHi, acc);

    float vals[8];
#pragma unroll
    for (int i = 0; i < 8; ++i) {
      vals[i] = __expf(acc[i] - mr[i]) * ir[i];
    }

    float* dst = att_t + (size_t)b * N_ * N_ + (size_t)(k0 + m) * N_ + q0 + 8 * hi;
    float4 lo4, hi4;
    lo4.x = vals[0]; lo4.y = vals[1]; lo4.z = vals[2]; lo4.w = vals[3];
    hi4.x = vals[4]; hi4.y = vals[5]; hi4.z = vals[6]; hi4.w = vals[7];
    *(float4*)(dst)     = lo4;
    *(float4*)(dst + 4) = hi4;
  }
}

// ---------------------------------------------------------------------------
// Kernel 3: out[b][c][q] = gamma * (sum_k v[b][c][k] * att_t[b][k][q]) + x[b][c][q]
// Per batch GEMM M=64(c), N=4096(q), K=4096(k). Block tile 64c x 32q, BK=32.
// Double-buffered LDS; staging via async global->LDS b128 copies (ASYNCcnt)
// issued for chunk i+1 before computing chunk i; one barrier per chunk.
// ---------------------------------------------------------------------------
__global__ void out_gemm_kernel(const float* __restrict__ vwp,
                                const float* __restrict__ att_t,
                                const float* __restrict__ x,
                                const float* __restrict__ gamma,
                                float* __restrict__ out) {
  __shared__ float Av[2][C_ * 32];   // [buf][c][kk], row stride 32
  __shared__ float Bt[2][32 * 64];   // [buf][kk][qq], row stride 64

  const int b     = blockIdx.y;
  const int qbase = blockIdx.x * 32;
  const int t     = threadIdx.x;
  const int wave  = t >> 5;
  const int lane  = t & 31;
  const int m     = lane & 15;
  const int hi    = lane >> 4;
  const int mt    = wave & 3;      // 4 c-tiles
  const int nt    = wave >> 2;     // 2 q-tiles
  const int cw    = mt * 16;
  const int qw0   = nt * 16;

  const float* vb = vwp   + (size_t)b * C_ * N_;
  const float* ab = att_t + (size_t)b * N_ * N_;

  // Per-thread staging coordinates (2 b128 chunks each for Av and Bt).
  const int arow0 = t >> 3,         acol0 = (t & 7) * 4;
  const int arow1 = (t + 256) >> 3, acol1 = ((t + 256) & 7) * 4;
  const int brow0 = t >> 4,         bcol0 = (t & 15) * 4;
  const int brow1 = (t + 256) >> 4, bcol1 = ((t + 256) & 15) * 4;

  auto stage = [&](int buf, int kc) {
    async_ld16(&Av[buf][arow0 * 32 + acol0], vb + (size_t)arow0 * N_ + kc + acol0);
    async_ld16(&Av[buf][arow1 * 32 + acol1], vb + (size_t)arow1 * N_ + kc + acol1);
    async_ld16(&Bt[buf][brow0 * 64 + bcol0], ab + (size_t)(kc + brow0) * N_ + qbase + bcol0);
    async_ld16(&Bt[buf][brow1 * 64 + bcol1], ab + (size_t)(kc + brow1) * N_ + qbase + bcol1);
  };

  v8f acc = {0.f, 0.f, 0.f, 0.f, 0.f, 0.f, 0.f, 0.f};

  // Prologue: stage chunk 0 into buffer 0.
  stage(0, 0);
  wait_async();
  __syncthreads();

  int cur = 0;
  for (int kc = 0; kc < N_; kc += 32) {
    const int kn = kc + 32;
    if (kn < N_) stage(cur ^ 1, kn);   // overlap next-chunk DMA with compute

    const float* __restrict__ Ac = &Av[cur][0];
    const float* __restrict__ Bc = &Bt[cur][0];
#pragma unroll
    for (int kk = 0; kk < 32; kk += 4) {
      v2f a, bfrag;
      const float2 af = *(const float2*)&Ac[(cw + m) * 32 + kk + 2 * hi];
      a.x = af.x;
      a.y = af.y;
      bfrag.x = Bc[(kk + 2 * hi) * 64 + qw0 + m];
      bfrag.y = Bc[(kk + 1 + 2 * hi) * 64 + qw0 + m];
      acc = wmma4(a, bfrag, acc);
    }

    wait_async();
    __syncthreads();
    cur ^= 1;
  }

  const float g  = gamma[0];
  const int   qg = qbase + qw0 + m;
  const float* xb = x   + (size_t)b * C_ * N_;
  float*       ob = out + (size_t)b * C_ * N_;
#pragma unroll
  for (int i = 0; i < 8; ++i) {
    const int    cg  = cw + i + 8 * hi;
    const size_t idx = (size_t)cg * N_ + qg;
    ob[idx] = g * acc[i] + xb[idx];
  }
}

// ---------------------------------------------------------------------------
// Launch
// ---------------------------------------------------------------------------
extern "C" void kernel_launch(void* const* d_in, const int* in_sizes, int n_in,
                              void* d_out, int out_size, void* d_ws, size_t ws_size,
                              hipStream_t stream) {
  const float* x     = (const float*)d_in[0];
  const float* wq    = (const float*)d_in[1];
  const float* bq    = (const float*)d_in[2];
  const float* wk    = (const float*)d_in[3];
  const float* bk    = (const float*)d_in[4];
  const float* wv    = (const float*)d_in[5];
  const float* bv    = (const float*)d_in[6];
  const float* gamma = (const float*)d_in[7];

  float* outp  = (float*)d_out;                       // B*C*N
  float* att_t = outp + (size_t)B_ * C_ * N_;         // B*N*N

  float* qwp    = (float*)d_ws;                       // B*CQ*N
  float* kwp    = qwp    + (size_t)B_ * CQ_ * N_;
  float* vwp    = kwp    + (size_t)B_ * CQ_ * N_;     // B*C*N
  float* rowm   = vwp    + (size_t)B_ * C_ * N_;      // B*N
  float* rowinv = rowm   + (size_t)B_ * N_;           // B*N

  qkv_kernel<<<dim3(B_ * N_ / 256), 256, 0, stream>>>(
      x, wq, bq, wk, bk, wv, bv, qwp, kwp, vwp);

  att_stats_kernel<<<dim3(N_ / 16, B_), 256, 0, stream>>>(qwp, kwp, rowm, rowinv);

  att_write_kernel<<<dim3(N_ / 16, N_ / 512, B_), 256, 0, stream>>>(
      qwp, kwp, rowm, rowinv, att_t);

  out_gemm_kernel<<<dim3(N_ / 32, B_), 256, 0, stream>>>(
      vwp, att_t, x, gamma, outp);
}